// EquivariantProductBasisBlock_65017214926999
// MI455X (gfx1250) — compile-verified
//
#include <hip/hip_runtime.h>

typedef __attribute__((ext_vector_type(2))) float v2f;
typedef __attribute__((ext_vector_type(8))) float v8f;

#define NNODES   65536
#define CDIM     128
#define BN       64        // nodes per block
#define LDSK     130       // padded K stride (bank-conflict-free, 8B aligned rows)
#define P0       5
#define P1       4

#define INV_SQ3   0.5773502691896258f
#define SQ2       1.4142135623730951f
#define SQ3       1.7320508075688772f
#define SQ35      0.7745966692414834f
#define INV_SQRT_C 0.08838834764831845f   // 1/sqrt(128)

__global__ __launch_bounds__(256)
void equi_product_fused(const float* __restrict__ feats,
                        const int*   __restrict__ species,
                        const float* __restrict__ sc,
                        const float* __restrict__ W0,
                        const float* __restrict__ W1,
                        const float* __restrict__ L0,
                        const float* __restrict__ L1,
                        float* __restrict__ out)
{
    extern __shared__ float smem[];   // [4][BN][LDSK]
    const int tid = threadIdx.x;
    const int node_base = blockIdx.x * BN;

    // Warm L2 with the L matrices (global_prefetch_b8)
    __builtin_prefetch(L0 + tid * 64, 0, 0);
    __builtin_prefetch(L1 + tid * 64, 0, 0);

    // ---------------- Phase 1: pointwise basis -> LDS ----------------
    #pragma unroll 4
    for (int i = 0; i < (BN * CDIM) / 256; ++i) {
        int idx = i * 256 + tid;
        int ln  = idx >> 7;          // local node
        int c   = idx & (CDIM - 1);  // channel
        int n   = node_base + ln;

        float4 f = ((const float4*)feats)[n * CDIM + c];
        float s = f.x, vx = f.y, vy = f.z, vz = f.w;
        float v2 = vx * vx + vy * vy + vz * vz;
        float s2 = s * s;

        int e = species[n];
        const float* w0 = W0 + (e * P0) * CDIM + c;   // stride CDIM per p
        const float* w1 = W1 + (e * P1) * CDIM + c;

        float o0 = w0[0] * s
                 + w0[CDIM] * s2
                 + w0[2 * CDIM] * (INV_SQ3 * v2)
                 + w0[3 * CDIM] * (s2 * s)
                 + w0[4 * CDIM] * (s * v2);
        float t1 = w1[0]
                 + w1[CDIM] * (SQ2 * s)
                 + w1[2 * CDIM] * (SQ3 * s2)
                 + w1[3 * CDIM] * (SQ35 * v2);

        smem[(0 * BN + ln) * LDSK + c] = o0;
        smem[(1 * BN + ln) * LDSK + c] = t1 * vx;
        smem[(2 * BN + ln) * LDSK + c] = t1 * vy;
        smem[(3 * BN + ln) * LDSK + c] = t1 * vz;
    }
    __syncthreads();

    // ---------------- Phase 2: fp32 WMMA GEMM against L0 / L1 ----------------
    const int wave = tid >> 5;          // 0..7  -> output-channel tile
    const int lane = tid & 31;
    const int half = lane >> 4;         // 0: lanes 0-15, 1: lanes 16-31
    const int l15  = lane & 15;
    const int n0   = wave * 16;         // output channel base

    // B fragments for both L matrices, resident in VGPRs for the whole block.
    // reg 2k+f : lane<16 -> L[4k+f][n0+l15], lane>=16 -> L[4k+2+f][n0+l15]
    float B0[64], B1[64];
    #pragma unroll
    for (int k = 0; k < 32; ++k) {
        #pragma unroll
        for (int fcomp = 0; fcomp < 2; ++fcomp) {
            int row = 4 * k + fcomp + 2 * half;
            B0[2 * k + fcomp] = L0[row * CDIM + n0 + l15];
            B1[2 * k + fcomp] = L1[row * CDIM + n0 + l15];
        }
    }

    #pragma unroll 1
    for (int m = 0; m < BN / 16; ++m) {
        v8f acc[4];
        // comp unrolled so (comp==0 ? B0 : B1) is compile-time and B stays in VGPRs
        #pragma unroll
        for (int comp = 0; comp < 4; ++comp) {
            const float* arow =
                smem + ((comp * BN + m * 16 + l15) * LDSK) + 2 * half;
            v8f a8 = {};
            acc[comp] = a8;
            #pragma unroll
            for (int k = 0; k < 32; ++k) {
                v2f a = *(const v2f*)(arow + 4 * k);     // ds_load_b64 (paired by compiler)
                v2f b;
                b.x = (comp == 0) ? B0[2 * k]     : B1[2 * k];
                b.y = (comp == 0) ? B0[2 * k + 1] : B1[2 * k + 1];
                acc[comp] = __builtin_amdgcn_wmma_f32_16x16x4_f32(
                        false, a, false, b, (short)0, acc[comp], false, false);
            }
        }
        // Epilogue: all four components together -> fully coalesced float4
        #pragma unroll
        for (int r = 0; r < 8; ++r) {
            int node = node_base + m * 16 + r + 8 * half;
            int off  = node * CDIM + n0 + l15;           // (node, d) index
            float4 s4 = ((const float4*)sc)[off];
            float4 o;
            o.x = acc[0][r] * INV_SQRT_C + s4.x;
            o.y = acc[1][r] * INV_SQRT_C + s4.y;
            o.z = acc[2][r] * INV_SQRT_C + s4.z;
            o.w = acc[3][r] * INV_SQRT_C + s4.w;
            ((float4*)out)[off] = o;
        }
    }
}

extern "C" void kernel_launch(void* const* d_in, const int* in_sizes, int n_in,
                              void* d_out, int out_size, void* d_ws, size_t ws_size,
                              hipStream_t stream) {
    const float* feats   = (const float*)d_in[0];
    const int*   species = (const int*)  d_in[1];
    const float* sc      = (const float*)d_in[2];
    const float* W0      = (const float*)d_in[3];
    const float* W1      = (const float*)d_in[4];
    const float* L0      = (const float*)d_in[5];
    const float* L1      = (const float*)d_in[6];
    float*       out     = (float*)d_out;

    dim3 grid(NNODES / BN);
    dim3 block(256);
    size_t lds_bytes = (size_t)4 * BN * LDSK * sizeof(float);   // 133,120 B
    hipLaunchKernelGGL(equi_product_fused, grid, block, lds_bytes, stream,
                       feats, species, sc, W0, W1, L0, L1, out);
}